// Net_15539191677077
// MI455X (gfx1250) — compile-verified
//
#include <hip/hip_runtime.h>

// ---------------- problem constants (match reference) ----------------
#define Gg    64
#define Nn    2048
#define DEG   8
#define NTOT  (Gg * Nn)          // 131072
#define Ee    (NTOT * DEG)       // 1048576
#define Kk    410                // ceil(0.2 * 2048)
#define NK    (Gg * Kk)          // 26240
#define Hh    3
#define Cc    20

typedef __attribute__((ext_vector_type(2))) float v2f;
typedef __attribute__((ext_vector_type(8))) float v8f;

// order-preserving float<->uint encoding for atomicMax on floats
__device__ __forceinline__ unsigned encf(float f) {
    unsigned b = __float_as_uint(f);
    return (b & 0x80000000u) ? ~b : (b | 0x80000000u);
}
__device__ __forceinline__ float decf(unsigned e) {
    unsigned b = (e & 0x80000000u) ? (e & 0x7FFFFFFFu) : ~e;
    return __uint_as_float(b);
}
__device__ __forceinline__ float leaky(float x) { return x > 0.f ? x : 0.2f * x; }

// ---------------- degree / dinv ----------------
__global__ void k_deg(const int* __restrict__ dst, float* __restrict__ deg) {
    int e = blockIdx.x * blockDim.x + threadIdx.x;
    if (e < Ee) atomicAdd(&deg[dst[e]], 1.0f);
}
__global__ void k_dinv(float* __restrict__ deg) {
    int i = blockIdx.x * blockDim.x + threadIdx.x;
    if (i < NTOT) deg[i] = rsqrtf(deg[i] + 1.0f);
}

// ---------------- h1 = x @ W1  (f32 WMMA 16x16x4, K=5 -> 2 chunks) ----------------
// Padding via clamped-address unconditional loads + select-to-zero: keeps the
// operand-load region branch-free so EXEC stays all-1s through the WMMAs.
__global__ void k_w1(const float* __restrict__ x, const float* __restrict__ W1,
                     float* __restrict__ h1) {
    int wave = (blockIdx.x * blockDim.x + threadIdx.x) >> 5;
    int lane = threadIdx.x & 31;
    int rowTile = wave << 4;           // NTOT divisible by 16 -> no guard
    int r  = lane & 15;
    int kh = lane >> 4;
    int row = rowTile + r;
    int rc = (r < 10) ? r : 9;         // clamped column for B loads
    v8f c = {};
#pragma unroll
    for (int kt = 0; kt < 2; kt++) {
        v2f a, b;
#pragma unroll
        for (int j = 0; j < 2; j++) {
            int k  = kt * 4 + kh * 2 + j;
            int kc = (k < 5) ? k : 4;
            float av = x[row * 5 + kc];
            float bv = W1[kc * 10 + rc];
            a[j] = (k < 5) ? av : 0.f;
            b[j] = (k < 5 && r < 10) ? bv : 0.f;
        }
        c = __builtin_amdgcn_wmma_f32_16x16x4_f32(false, a, false, b, (short)0, c,
                                                  false, false);
    }
    if (r < 10) {
#pragma unroll
        for (int rr = 0; rr < 8; rr++)
            h1[(rowTile + rr + 8 * kh) * 10 + r] = c[rr];
    }
}

// ---------------- GCN1 aggregate + finalize ----------------
__global__ void k_agg10(const int* __restrict__ src, const int* __restrict__ dst,
                        const float* __restrict__ dinv, const float* __restrict__ h1,
                        float* __restrict__ x1) {
    int e = blockIdx.x * blockDim.x + threadIdx.x;
    if (e >= Ee) return;
    int s = src[e], d = dst[e];
    float coef = dinv[s] * dinv[d];
#pragma unroll
    for (int cch = 0; cch < 10; cch++)
        atomicAdd(&x1[d * 10 + cch], h1[s * 10 + cch] * coef);
}
__global__ void k_fin10(const float* __restrict__ h1, const float* __restrict__ dinv,
                        const float* __restrict__ b1, float* __restrict__ x1) {
    int i = blockIdx.x * blockDim.x + threadIdx.x;
    if (i >= NTOT) return;
    float di2 = dinv[i] * dinv[i];
#pragma unroll
    for (int cch = 0; cch < 10; cch++)
        x1[i * 10 + cch] = x1[i * 10 + cch] + h1[i * 10 + cch] * di2 + b1[cch];
}

// ---------------- scorer GCN ----------------
__global__ void k_hs(const float* __restrict__ x1, const float* __restrict__ Ws,
                     float* __restrict__ hs) {
    int i = blockIdx.x * blockDim.x + threadIdx.x;
    if (i >= NTOT) return;
    float s = 0.f;
#pragma unroll
    for (int cch = 0; cch < 10; cch++) s += x1[i * 10 + cch] * Ws[cch];
    hs[i] = s;
}
__global__ void k_agg1(const int* __restrict__ src, const int* __restrict__ dst,
                       const float* __restrict__ dinv, const float* __restrict__ hs,
                       float* __restrict__ score) {
    int e = blockIdx.x * blockDim.x + threadIdx.x;
    if (e >= Ee) return;
    int s = src[e], d = dst[e];
    atomicAdd(&score[d], hs[s] * dinv[s] * dinv[d]);
}
__global__ void k_fin1(const float* __restrict__ hs, const float* __restrict__ dinv,
                       const float* __restrict__ bs, float* __restrict__ score) {
    int i = blockIdx.x * blockDim.x + threadIdx.x;
    if (i >= NTOT) return;
    score[i] = score[i] + hs[i] * dinv[i] * dinv[i] + bs[0];
}

// ---------------- per-graph top-K via bitonic sort in LDS ----------------
__global__ void k_topk(const float* __restrict__ score, const float* __restrict__ x1,
                       float* __restrict__ xp, int* __restrict__ perm,
                       int* __restrict__ newidx) {
    __shared__ float sv[Nn];
    __shared__ int   si[Nn];
    int g = blockIdx.x;
    for (int i = threadIdx.x; i < Nn; i += blockDim.x) {
        sv[i] = score[g * Nn + i];
        si[i] = i;
    }
    __syncthreads();
    for (int k = 2; k <= Nn; k <<= 1) {
        for (int j = k >> 1; j > 0; j >>= 1) {
            for (int t = threadIdx.x; t < Nn; t += blockDim.x) {
                int ixj = t ^ j;
                if (ixj > t) {
                    float a = sv[t], b = sv[ixj];
                    bool up = ((t & k) == 0);      // descending sort
                    bool sw = up ? (a < b) : (a > b);
                    if (sw) {
                        sv[t] = b; sv[ixj] = a;
                        int tmp = si[t]; si[t] = si[ixj]; si[ixj] = tmp;
                    }
                }
            }
            __syncthreads();
        }
    }
    for (int t = threadIdx.x; t < Kk; t += blockDim.x) {
        int node = g * Nn + si[t];
        int p = g * Kk + t;
        perm[p] = node;
        newidx[node] = p;
        float gate = tanhf(sv[t]);
#pragma unroll
        for (int cch = 0; cch < 10; cch++)
            xp[p * 10 + cch] = x1[node * 10 + cch] * gate;
    }
}

// ---------------- hg = xp @ Wg  (f32 WMMA, K=10 -> 3 chunks, 60 cols -> 4 tiles) ----
__global__ void k_wg(const float* __restrict__ xp, const float* __restrict__ Wg,
                     float* __restrict__ hg) {
    int wave = (blockIdx.x * blockDim.x + threadIdx.x) >> 5;
    int lane = threadIdx.x & 31;
    int rowTile = wave << 4;           // NK divisible by 16
    int r  = lane & 15;
    int kh = lane >> 4;
    int row = rowTile + r;
    v2f a[3];
#pragma unroll
    for (int kt = 0; kt < 3; kt++)
#pragma unroll
        for (int j = 0; j < 2; j++) {
            int k  = kt * 4 + kh * 2 + j;
            int kc = (k < 10) ? k : 9;
            float av = xp[row * 10 + kc];
            a[kt][j] = (k < 10) ? av : 0.f;
        }
#pragma unroll
    for (int nt = 0; nt < 4; nt++) {
        int n  = nt * 16 + r;
        int nc = (n < 60) ? n : 59;
        v8f c = {};
#pragma unroll
        for (int kt = 0; kt < 3; kt++) {
            v2f b;
#pragma unroll
            for (int j = 0; j < 2; j++) {
                int k  = kt * 4 + kh * 2 + j;
                int kc = (k < 10) ? k : 9;
                float bv = Wg[kc * 60 + nc];
                b[j] = (k < 10 && n < 60) ? bv : 0.f;
            }
            c = __builtin_amdgcn_wmma_f32_16x16x4_f32(false, a[kt], false, b, (short)0,
                                                      c, false, false);
        }
        if (n < 60) {
#pragma unroll
            for (int rr = 0; rr < 8; rr++)
                hg[(rowTile + rr + 8 * kh) * 60 + n] = c[rr];
        }
    }
}

// ---------------- GAT attention coefficients ----------------
__global__ void k_att(const float* __restrict__ hg, const float* __restrict__ a_src,
                      const float* __restrict__ a_dst, float* __restrict__ als,
                      float* __restrict__ ald) {
    int p = blockIdx.x * blockDim.x + threadIdx.x;
    if (p >= NK) return;
#pragma unroll
    for (int h = 0; h < Hh; h++) {
        float s1 = 0.f, s2 = 0.f;
#pragma unroll
        for (int cch = 0; cch < Cc; cch++) {
            float v = hg[p * 60 + h * Cc + cch];
            s1 += v * a_src[h * Cc + cch];
            s2 += v * a_dst[h * Cc + cch];
        }
        als[p * Hh + h] = s1;
        ald[p * Hh + h] = s2;
    }
}

// ---------------- edge softmax: max / sum / accumulate ----------------
__global__ void k_max_e(const int* __restrict__ src, const int* __restrict__ dst,
                        const int* __restrict__ newidx, const float* __restrict__ als,
                        const float* __restrict__ ald, unsigned* __restrict__ mx) {
    int e = blockIdx.x * blockDim.x + threadIdx.x;
    if (e >= Ee) return;
    int s = newidx[src[e]], d = newidx[dst[e]];
    if (s < 0 || d < 0) return;
#pragma unroll
    for (int h = 0; h < Hh; h++)
        atomicMax(&mx[d * Hh + h], encf(leaky(als[s * Hh + h] + ald[d * Hh + h])));
}
__global__ void k_max_s(const float* __restrict__ als, const float* __restrict__ ald,
                        unsigned* __restrict__ mx) {
    int p = blockIdx.x * blockDim.x + threadIdx.x;
    if (p >= NK) return;
#pragma unroll
    for (int h = 0; h < Hh; h++)
        atomicMax(&mx[p * Hh + h], encf(leaky(als[p * Hh + h] + ald[p * Hh + h])));
}
__global__ void k_sum_e(const int* __restrict__ src, const int* __restrict__ dst,
                        const int* __restrict__ newidx, const float* __restrict__ als,
                        const float* __restrict__ ald, const unsigned* __restrict__ mx,
                        float* __restrict__ denom) {
    int e = blockIdx.x * blockDim.x + threadIdx.x;
    if (e >= Ee) return;
    int s = newidx[src[e]], d = newidx[dst[e]];
    if (s < 0 || d < 0) return;
#pragma unroll
    for (int h = 0; h < Hh; h++) {
        float lg = leaky(als[s * Hh + h] + ald[d * Hh + h]);
        atomicAdd(&denom[d * Hh + h], expf(lg - decf(mx[d * Hh + h])));
    }
}
__global__ void k_sum_s(const float* __restrict__ als, const float* __restrict__ ald,
                        const unsigned* __restrict__ mx, float* __restrict__ denom) {
    int p = blockIdx.x * blockDim.x + threadIdx.x;
    if (p >= NK) return;
#pragma unroll
    for (int h = 0; h < Hh; h++) {
        float lg = leaky(als[p * Hh + h] + ald[p * Hh + h]);
        atomicAdd(&denom[p * Hh + h], expf(lg - decf(mx[p * Hh + h])));
    }
}
__global__ void k_acc_e(const int* __restrict__ src, const int* __restrict__ dst,
                        const int* __restrict__ newidx, const float* __restrict__ als,
                        const float* __restrict__ ald, const unsigned* __restrict__ mx,
                        const float* __restrict__ denom, const float* __restrict__ hg,
                        float* __restrict__ outacc) {
    int e = blockIdx.x * blockDim.x + threadIdx.x;
    if (e >= Ee) return;
    int s = newidx[src[e]], d = newidx[dst[e]];
    if (s < 0 || d < 0) return;
#pragma unroll
    for (int h = 0; h < Hh; h++) {
        float lg = leaky(als[s * Hh + h] + ald[d * Hh + h]);
        float alpha = expf(lg - decf(mx[d * Hh + h])) / denom[d * Hh + h];
#pragma unroll
        for (int cch = 0; cch < Cc; cch++)
            atomicAdd(&outacc[d * 60 + h * Cc + cch],
                      hg[s * 60 + h * Cc + cch] * alpha);
    }
}
__global__ void k_acc_s(const float* __restrict__ als, const float* __restrict__ ald,
                        const unsigned* __restrict__ mx, const float* __restrict__ denom,
                        const float* __restrict__ hg, float* __restrict__ outacc) {
    int p = blockIdx.x * blockDim.x + threadIdx.x;
    if (p >= NK) return;
#pragma unroll
    for (int h = 0; h < Hh; h++) {
        float lg = leaky(als[p * Hh + h] + ald[p * Hh + h]);
        float alpha = expf(lg - decf(mx[p * Hh + h])) / denom[p * Hh + h];
#pragma unroll
        for (int cch = 0; cch < Cc; cch++)
            atomicAdd(&outacc[p * 60 + h * Cc + cch],
                      hg[p * 60 + h * Cc + cch] * alpha);
    }
}

// ---------------- readout: g[graph] += outacc[p] + bg ----------------
__global__ void k_readout(const float* __restrict__ outacc, const float* __restrict__ bg,
                          float* __restrict__ gbuf) {
    int idx = blockIdx.x * blockDim.x + threadIdx.x;
    if (idx >= NK * 60) return;
    int p = idx / 60, j = idx % 60;
    atomicAdd(&gbuf[(p / Kk) * 60 + j], outacc[idx] + bg[j]);
}

// ---------------- final MLP + log_softmax ----------------
__global__ void k_mlp(const float* __restrict__ gbuf, const float* __restrict__ Wf1,
                      const float* __restrict__ bf1, const float* __restrict__ Wf2,
                      const float* __restrict__ bf2, float* __restrict__ out) {
    int gr = blockIdx.x * blockDim.x + threadIdx.x;
    if (gr >= Gg) return;
    float hid[30];
#pragma unroll 1
    for (int j = 0; j < 30; j++) {
        float s = bf1[j];
        for (int i = 0; i < 60; i++) s += gbuf[gr * 60 + i] * Wf1[i * 30 + j];
        hid[j] = fmaxf(s, 0.f);
    }
    float z[3];
#pragma unroll
    for (int j = 0; j < 3; j++) {
        float s = bf2[j];
        for (int i = 0; i < 30; i++) s += hid[i] * Wf2[i * 3 + j];
        z[j] = s;
    }
    float m = fmaxf(z[0], fmaxf(z[1], z[2]));
    float den = expf(z[0] - m) + expf(z[1] - m) + expf(z[2] - m);
    float lg = logf(den);
#pragma unroll
    for (int j = 0; j < 3; j++) out[gr * 3 + j] = z[j] - m - lg;
}

// ---------------- host side ----------------
extern "C" void kernel_launch(void* const* d_in, const int* in_sizes, int n_in,
                              void* d_out, int out_size, void* d_ws, size_t ws_size,
                              hipStream_t stream) {
    const float* x     = (const float*)d_in[0];
    const int*   src   = (const int*)d_in[1];
    const int*   dst   = (const int*)d_in[2];
    // d_in[3] = batch (unused: pooled node p belongs to graph p/K by construction)
    const float* W1    = (const float*)d_in[4];
    const float* b1    = (const float*)d_in[5];
    const float* Ws    = (const float*)d_in[6];
    const float* bs    = (const float*)d_in[7];
    const float* Wg    = (const float*)d_in[8];
    const float* a_src = (const float*)d_in[9];
    const float* a_dst = (const float*)d_in[10];
    const float* bg    = (const float*)d_in[11];
    const float* Wf1   = (const float*)d_in[12];
    const float* bf1   = (const float*)d_in[13];
    const float* Wf2   = (const float*)d_in[14];
    const float* bf2   = (const float*)d_in[15];
    float* out = (float*)d_out;

    // workspace layout (floats/ints, all 4-byte)
    float*    dinv   = (float*)d_ws;              // NTOT
    float*    h1     = dinv + NTOT;               // NTOT*10
    float*    x1     = h1 + (size_t)NTOT * 10;    // NTOT*10
    float*    hs     = x1 + (size_t)NTOT * 10;    // NTOT
    float*    score  = hs + NTOT;                 // NTOT
    int*      newidx = (int*)(score + NTOT);      // NTOT
    int*      perm   = newidx + NTOT;             // NK
    float*    xp     = (float*)(perm + NK);       // NK*10
    float*    hg     = xp + (size_t)NK * 10;      // NK*60
    float*    als    = hg + (size_t)NK * 60;      // NK*3
    float*    ald    = als + (size_t)NK * 3;      // NK*3
    unsigned* mx     = (unsigned*)(ald + (size_t)NK * 3); // NK*3
    float*    denom  = (float*)(mx + (size_t)NK * 3);     // NK*3
    float*    outacc = denom + (size_t)NK * 3;    // NK*60
    float*    gbuf   = outacc + (size_t)NK * 60;  // G*60
    size_t totalElems = (size_t)NTOT * 24 + (size_t)NK * 143 + (size_t)Gg * 60;

    // zero all accumulators (enc(0) is below every finite float for the max pass)
    hipMemsetAsync(d_ws, 0, totalElems * 4, stream);
    // new_idx = -1
    hipMemsetAsync(newidx, 0xFF, (size_t)NTOT * 4, stream);

    const int TB = 256;
    dim3 blk(TB);
    // degree + dinv
    k_deg <<<(Ee + TB - 1) / TB, blk, 0, stream>>>(dst, dinv);
    k_w1  <<<NTOT / 16 / (TB / 32), blk, 0, stream>>>(x, W1, h1);  // 1024 blocks
    k_dinv<<<(NTOT + TB - 1) / TB, blk, 0, stream>>>(dinv);
    // GCN1
    k_agg10<<<(Ee + TB - 1) / TB, blk, 0, stream>>>(src, dst, dinv, h1, x1);
    k_fin10<<<(NTOT + TB - 1) / TB, blk, 0, stream>>>(h1, dinv, b1, x1);
    // scorer GCN
    k_hs  <<<(NTOT + TB - 1) / TB, blk, 0, stream>>>(x1, Ws, hs);
    k_agg1<<<(Ee + TB - 1) / TB, blk, 0, stream>>>(src, dst, dinv, hs, score);
    k_fin1<<<(NTOT + TB - 1) / TB, blk, 0, stream>>>(hs, dinv, bs, score);
    // SAGPool top-k (one workgroup per graph)
    k_topk<<<Gg, 1024, 0, stream>>>(score, x1, xp, perm, newidx);
    // GAT linear transform
    k_wg  <<<NK / 16 / (TB / 32), blk, 0, stream>>>(xp, Wg, hg);   // 205 blocks
    k_att <<<(NK + TB - 1) / TB, blk, 0, stream>>>(hg, a_src, a_dst, als, ald);
    // edge softmax (3 passes; edges + self loops)
    k_max_e<<<(Ee + TB - 1) / TB, blk, 0, stream>>>(src, dst, newidx, als, ald, mx);
    k_max_s<<<(NK + TB - 1) / TB, blk, 0, stream>>>(als, ald, mx);
    k_sum_e<<<(Ee + TB - 1) / TB, blk, 0, stream>>>(src, dst, newidx, als, ald, mx, denom);
    k_sum_s<<<(NK + TB - 1) / TB, blk, 0, stream>>>(als, ald, mx, denom);
    k_acc_e<<<(Ee + TB - 1) / TB, blk, 0, stream>>>(src, dst, newidx, als, ald, mx,
                                                    denom, hg, outacc);
    k_acc_s<<<(NK + TB - 1) / TB, blk, 0, stream>>>(als, ald, mx, denom, hg, outacc);
    // readout + MLP
    k_readout<<<((size_t)NK * 60 + TB - 1) / TB, blk, 0, stream>>>(outacc, bg, gbuf);
    k_mlp  <<<1, 64, 0, stream>>>(gbuf, Wf1, bf1, Wf2, bf2, out);
    (void)in_sizes; (void)n_in; (void)out_size; (void)ws_size;
}